// HashGridEncoder_44495861186615
// MI455X (gfx1250) — compile-verified
//
#include <hip/hip_runtime.h>

// Instant-NGP hash grid encode (no interpolation): gather-bound, no matmul.
// Strategy: LDS-stage x via gfx1250 async global->LDS (ASYNCcnt path), keep
// the 64MB tables L2-resident by using non-temporal hints for all streamed
// traffic, fully unroll the 16-level loop for MLP, NT b128 output stores.
// Full blocks take an unguarded staging path so the async copies lower to
// saddr + scale_offset + imm-offset form with no EXEC branching.

#define LEVELS 16
#define HASHMAP_SIZE 524288
#define HASH_MASK (HASHMAP_SIZE - 1)
#define BLOCK 256

typedef __attribute__((ext_vector_type(2))) float f2;
typedef __attribute__((ext_vector_type(4))) float f4;

typedef __attribute__((address_space(1))) int gint_t;
typedef __attribute__((address_space(3))) int lint_t;

#if defined(__has_builtin)
#if __has_builtin(__builtin_amdgcn_global_load_async_to_lds_b32) && \
    __has_builtin(__builtin_amdgcn_s_wait_asynccnt)
#define USE_ASYNC_LDS 1
#endif
#endif

__device__ __forceinline__ void stage_one(const float* gx, float* sx_elem) {
#ifdef USE_ASYNC_LDS
    __builtin_amdgcn_global_load_async_to_lds_b32(
        (gint_t*)(float*)gx, (lint_t*)sx_elem, 0, 0);
#else
    *sx_elem = __builtin_nontemporal_load(gx);
#endif
}

__global__ __launch_bounds__(BLOCK) void hashgrid_encode_kernel(
    const float* __restrict__ x, const float* __restrict__ tables,
    float* __restrict__ out, int npts)
{
    // res_l = int(16 * 1.5^l), exact in fp32
    constexpr float kRes[LEVELS] = {
        16.f, 24.f, 36.f, 54.f, 81.f, 121.f, 182.f, 273.f,
        410.f, 615.f, 922.f, 1383.f, 2075.f, 3113.f, 4670.f, 7006.f};

    __shared__ float sx[BLOCK * 3];

    const int t = threadIdx.x;
    const long long base = (long long)blockIdx.x * BLOCK;
    const float* gx = x + base * 3;
    const bool full = (base + BLOCK) <= (long long)npts;  // block-uniform

    if (full) {
        // Common case: no predication -> clean async b32 copies.
        #pragma unroll
        for (int j = 0; j < 3; ++j) {
            const int e = j * BLOCK + t;
            stage_one(gx + e, &sx[e]);
        }
    } else {
        const int valid3 = (int)((long long)npts * 3 - base * 3);
        #pragma unroll
        for (int j = 0; j < 3; ++j) {
            const int e = j * BLOCK + t;
            if (e < valid3) stage_one(gx + e, &sx[e]);
        }
    }
#ifdef USE_ASYNC_LDS
    __builtin_amdgcn_s_wait_asynccnt(0);
#endif
    __syncthreads();

    const long long n = base + t;
    if (n >= npts) return;

    // Stride-3 LDS reads: 3 and 64 banks are coprime -> conflict-free.
    const float fx = sx[3 * t + 0];
    const float fy = sx[3 * t + 1];
    const float fz = sx[3 * t + 2];

    float r[2 * LEVELS];
    #pragma unroll
    for (int l = 0; l < LEVELS; ++l) {
        const float s = kRes[l];
        const int p0 = (int)floorf(fx * s);
        const int p1 = (int)floorf(fy * s);
        const int p2 = (int)floorf(fz * s);
        // int32 wraparound mul + xor; mod 2^19 (sign of divisor) == & mask
        const unsigned h = (unsigned)p0 * 73856093u
                         ^ (unsigned)p1 * 19349663u
                         ^ (unsigned)p2 * 83492791u;
        const unsigned idx = h & HASH_MASK;
        // 8B gather; tables stay L2-resident (64MB << 192MB L2), default RT hint
        const f2 v = *(const f2*)(tables + ((size_t)l * HASHMAP_SIZE + idx) * 2);
        r[2 * l + 0] = v.x;
        r[2 * l + 1] = v.y;
    }

    // 128 bytes per point, streamed once: non-temporal b128 stores so the
    // 256MB output never evicts the hash tables from L2.
    float* op = out + n * (2 * LEVELS);
    #pragma unroll
    for (int i = 0; i < 8; ++i) {
        f4 w = { r[4 * i + 0], r[4 * i + 1], r[4 * i + 2], r[4 * i + 3] };
        __builtin_nontemporal_store(w, (f4*)(op + 4 * i));
    }
}

extern "C" void kernel_launch(void* const* d_in, const int* in_sizes, int n_in,
                              void* d_out, int out_size, void* d_ws, size_t ws_size,
                              hipStream_t stream) {
    const float* x      = (const float*)d_in[0];   // [N,3] f32
    const float* tables = (const float*)d_in[1];   // [16,524288,2] f32
    float* out          = (float*)d_out;           // [N,32] f32

    const int npts = in_sizes[0] / 3;
    const int grid = (npts + BLOCK - 1) / BLOCK;
    hipLaunchKernelGGL(hashgrid_encode_kernel, dim3(grid), dim3(BLOCK), 0, stream,
                       x, tables, out, npts);
}